// ItemEmbeddingLayer_48971217109156
// MI455X (gfx1250) — compile-verified
//
#include <hip/hip_runtime.h>
#include <hip/hip_bf16.h>

// Fused double-gather + concat:
//   out[row, 0:128]   = item_embedding[idx[row], :]
//   out[row, 128:146] = genre_table[idx[row], :]
//
// MI455X strategy (memory-bound, 0 FLOPs -> WMMA inapplicable):
//  - tables (58.4 MB) fit in 192 MB L2 -> RT loads keep them resident
//  - output (612 MB write-once) streamed with TH=NT stores (no L2 pollution)
//  - wave32 chunking: 32 consecutive rows per wave; indices loaded with ONE
//    coalesced b32 load, broadcast per-row via v_readlane -> SGPR, so all
//    gather/store addressing is SADDR-form (scalar base + lane offset)
//  - global_prefetch_b8 two rows ahead to hide index->row latency

typedef __attribute__((ext_vector_type(2))) float v2f;
typedef __attribute__((ext_vector_type(4))) float v4f;

#define EMB_W   128   // embedding row width (floats)
#define GEN_W   18    // genre row width (floats)
#define OUT_W   146   // output row width (floats)

__global__ __launch_bounds__(256) void ItemEmbeddingLayer_gather_concat(
    const int*   __restrict__ idxs,   // [batch]
    const float* __restrict__ emb,    // [NUM_ITEMS, 128]
    const float* __restrict__ gen,    // [NUM_ITEMS, 18]
    float*       __restrict__ out,    // [batch, 146]
    int batch)
{
  const int lane = (int)(threadIdx.x & 31);
  const int wid  = (int)((blockIdx.x * blockDim.x + threadIdx.x) >> 5);
  const int nw   = (int)((gridDim.x * blockDim.x) >> 5);

  for (int base = wid * 32; base < batch; base += nw * 32) {
    // One coalesced 128B load fetches this wave's 32 row indices.
    const int myrow = base + lane;
    int vidx = 0;
    if (myrow < batch) vidx = idxs[myrow];

    int rcnt = batch - base;
    if (rcnt > 32) rcnt = 32;

#pragma unroll 4
    for (int r = 0; r < rcnt; ++r) {
      const int idx = __builtin_amdgcn_readlane(vidx, r);  // SGPR-uniform index
      const int row = base + r;

      // Prefetch the gather rows 2 iterations ahead (global_prefetch_b8):
      // 512B emb row = 4 x 128B cachelines, genre row = 1 cacheline.
      if (r + 2 < rcnt) {
        const int pidx = __builtin_amdgcn_readlane(vidx, r + 2);
        if (lane < 4)
          __builtin_prefetch(emb + (size_t)pidx * EMB_W + (lane << 5), 0, 3);
        if (lane == 4)
          __builtin_prefetch(gen + (size_t)pidx * GEN_W, 0, 3);
      }

      // ---- gather (SADDR base is scalar since idx is uniform) ----
      const v4f* erow = (const v4f*)(emb + (size_t)idx * EMB_W); // 512B-aligned row
      const v2f* grow = (const v2f*)(gen + (size_t)idx * GEN_W); // 8B-aligned row

      v4f e = erow[lane];             // global_load_b128, 512B/wave coalesced
      v2f g;
      const bool has_g = (lane < 9);  // 9 x float2 = 18 floats
      if (has_g) g = grow[lane];

      // ---- streaming (non-temporal) store of the concatenated row ----
      // Row base = row*584 bytes: 8B-aligned, so store as v2f.
      v2f* orow = (v2f*)(out + (size_t)row * OUT_W);
      v2f lo = {e.x, e.y};
      v2f hi = {e.z, e.w};
      __builtin_nontemporal_store(lo, orow + 2 * lane);     // floats [0,128)
      __builtin_nontemporal_store(hi, orow + 2 * lane + 1);
      if (has_g)
        __builtin_nontemporal_store(g, orow + 64 + lane);   // floats [128,146)
    }
  }
}

extern "C" void kernel_launch(void* const* d_in, const int* in_sizes, int n_in,
                              void* d_out, int out_size, void* d_ws, size_t ws_size,
                              hipStream_t stream) {
  const int*   idxs = (const int*)d_in[0];     // item_inputs    [batch]
  const float* emb  = (const float*)d_in[1];   // item_embedding [NUM_ITEMS,128]
  const float* gen  = (const float*)d_in[2];   // genre_table    [NUM_ITEMS,18]
  float*       out  = (float*)d_out;           // [batch, 146]

  const int batch = in_sizes[0];

  // ceil(batch/32) waves, 8 wave32 per 256-thread block.
  const int threads = 256;
  long long waves = ((long long)batch + 31) / 32;
  long long blk   = (waves + 7) / 8;
  if (blk > 8192) blk = 8192;   // grid-stride covers the rest
  if (blk < 1)    blk = 1;

  ItemEmbeddingLayer_gather_concat<<<(int)blk, threads, 0, stream>>>(
      idxs, emb, gen, out, batch);
}